// MultiHeadAttention_738734375358
// MI455X (gfx1250) — compile-verified
//
#include <hip/hip_runtime.h>

#define B_SZ 2
#define N_SZ 2048
#define M_SZ 2048
#define C_SZ 1024
#define H_SZ 16
#define HD_SZ 64
#define SCALE_F 0.125f   // HD^-0.5

typedef float v2f __attribute__((ext_vector_type(2)));
typedef float v8f __attribute__((ext_vector_type(8)));

__device__ __forceinline__ v8f wmma4(v2f a, v2f b, v8f c) {
  // D = A(16x4) * B(4x16) + C, all f32.  args: (neg_a, A, neg_b, B, c_mod, C, reuse_a, reuse_b)
  return __builtin_amdgcn_wmma_f32_16x16x4_f32(false, a, false, b, (short)0, c, false, false);
}

// ---------------------------------------------------------------------------
// Generic WMMA GEMM: Out = X(R x K) @ W(K x S) + bias, 32x64 tile per wave.
// mode 0: write to (B,H,rowsPerBatch,64) layout (q)
// mode 1: cols [0,C) -> outA, cols [C,2C) -> outB, both (B,H,rowsPerBatch,64) (k,v)
// mode 2: plain row-major (out projection)
// ---------------------------------------------------------------------------
__global__ __launch_bounds__(256) void gemm_wmma_kernel(
    const float* __restrict__ X, const float* __restrict__ W,
    const float* __restrict__ bias,
    float* __restrict__ outA, float* __restrict__ outB,
    int K, int S, int rowsPerBatch, int mode)
{
  const int lane   = threadIdx.x & 31;
  const int laneLo = lane & 15;
  const int laneHi = lane >> 4;
  const int gw     = blockIdx.x * 8 + (threadIdx.x >> 5);
  const int ctiles = S >> 6;
  const int r0 = (gw / ctiles) << 5;
  const int c0 = (gw % ctiles) << 6;

  const v8f z = {0.f,0.f,0.f,0.f,0.f,0.f,0.f,0.f};
  v8f acc[2][4];
#pragma unroll
  for (int i = 0; i < 2; ++i)
#pragma unroll
    for (int j = 0; j < 4; ++j) acc[i][j] = z;

  const float* xp0 = X + (size_t)(r0 + laneLo) * K + 2 * laneHi;
  const float* xp1 = xp0 + (size_t)16 * K;
  const float* wp0 = W + (size_t)(2 * laneHi) * S + c0 + laneLo;

  for (int k = 0; k < K; k += 4) {
    v2f a0 = *(const v2f*)(xp0 + k);
    v2f a1 = *(const v2f*)(xp1 + k);
    const float* wk = wp0 + (size_t)k * S;
#pragma unroll
    for (int j = 0; j < 4; ++j) {
      v2f bf;
      bf.x = wk[j * 16];
      bf.y = wk[j * 16 + S];
      acc[0][j] = wmma4(a0, bf, acc[0][j]);
      acc[1][j] = wmma4(a1, bf, acc[1][j]);
    }
  }

#pragma unroll
  for (int i = 0; i < 2; ++i)
#pragma unroll
    for (int j = 0; j < 4; ++j)
#pragma unroll
      for (int v = 0; v < 8; ++v) {
        const int r = r0 + 16 * i + v + 8 * laneHi;
        const int c = c0 + 16 * j + laneLo;
        const float val = acc[i][j][v] + bias[c];
        if (mode == 2) {
          outA[(size_t)r * S + c] = val;
        } else {
          const int b = r / rowsPerBatch;
          const int n = r - b * rowsPerBatch;
          int cc = c;
          float* dst = outA;
          if (mode == 1 && cc >= C_SZ) { dst = outB; cc -= C_SZ; }
          const int h = cc >> 6;
          const int d = cc & 63;
          dst[(((size_t)b * H_SZ + h) * rowsPerBatch + n) * HD_SZ + d] = val;
        }
      }
}

// ---------------------------------------------------------------------------
// Attention: one block per (b, 16-query tile); one wave per head.
// Pass 1: streaming online max/sum.  Pass 2: recompute S, write normalized
// probs (via LDS, coalesced (m,h)-contiguous stores) and accumulate O = P@V.
// ---------------------------------------------------------------------------
#define CHUNK 32
#define PSTRIDE (CHUNK * H_SZ + 4)   // 516 dwords: bank rotation of 4/row

__global__ __launch_bounds__(512) void attn_wmma_kernel(
    const float* __restrict__ Q,    // (B,H,N,64)
    const float* __restrict__ Kk,   // (B,H,M,64)
    const float* __restrict__ Vv,   // (B,H,M,64)
    float* __restrict__ attn,       // (B,N,M,H)
    float* __restrict__ Ohead)      // (B,N,H,64)
{
  __shared__ __align__(16) float p_lds[16 * PSTRIDE];

  const int lane   = threadIdx.x & 31;
  const int laneLo = lane & 15;
  const int laneHi = lane >> 4;
  const int h      = threadIdx.x >> 5;                 // head = wave id
  const int b      = blockIdx.x / (N_SZ / 16);
  const int n0     = (blockIdx.x % (N_SZ / 16)) * 16;

  const float* qbase = Q  + (((size_t)b * H_SZ + h) * N_SZ + n0) * HD_SZ;
  const float* kbase = Kk + ((size_t)b * H_SZ + h) * (size_t)M_SZ * HD_SZ;
  const float* vbase = Vv + ((size_t)b * H_SZ + h) * (size_t)M_SZ * HD_SZ;

  // Q A-fragments for all 16 k-steps, pre-scaled
  v2f qf[16];
#pragma unroll
  for (int ks = 0; ks < 16; ++ks) {
    const float* p = qbase + (size_t)laneLo * HD_SZ + 4 * ks + 2 * laneHi;
    qf[ks].x = p[0] * SCALE_F;
    qf[ks].y = p[1] * SCALE_F;
  }

  const v8f z = {0.f,0.f,0.f,0.f,0.f,0.f,0.f,0.f};

  // ---- pass 1: online row max / sum-exp over all M ----
  float rm[8], rs[8];
#pragma unroll
  for (int v = 0; v < 8; ++v) { rm[v] = -3.0e38f; rs[v] = 0.f; }

  for (int m0 = 0; m0 < M_SZ; m0 += 16) {
    v8f s = z;
    const float* kp = kbase + (size_t)(m0 + laneLo) * HD_SZ + 2 * laneHi;
#pragma unroll
    for (int ks = 0; ks < 16; ++ks) {
      v2f kf = *(const v2f*)(kp + 4 * ks);
      s = wmma4(qf[ks], kf, s);
    }
#pragma unroll
    for (int v = 0; v < 8; ++v) {
      float x = s[v];
      float tm = x;
      tm = fmaxf(tm, __shfl_xor(tm, 1));
      tm = fmaxf(tm, __shfl_xor(tm, 2));
      tm = fmaxf(tm, __shfl_xor(tm, 4));
      tm = fmaxf(tm, __shfl_xor(tm, 8));   // rows stay separated per 16-lane half
      const float nm = fmaxf(rm[v], tm);
      float ts = __expf(x - nm);
      ts += __shfl_xor(ts, 1);
      ts += __shfl_xor(ts, 2);
      ts += __shfl_xor(ts, 4);
      ts += __shfl_xor(ts, 8);
      rs[v] = rs[v] * __expf(rm[v] - nm) + ts;
      rm[v] = nm;
    }
  }
  float rinv[8];
#pragma unroll
  for (int v = 0; v < 8; ++v) rinv[v] = 1.0f / rs[v];

  // ---- pass 2: recompute S, emit normalized probs, accumulate O = P@V ----
  v8f oacc[4];
#pragma unroll
  for (int j = 0; j < 4; ++j) oacc[j] = z;

  for (int m0 = 0; m0 < M_SZ; m0 += CHUNK) {
#pragma unroll
    for (int t = 0; t < CHUNK / 16; ++t) {
      const int mt = m0 + t * 16;
      v8f s = z;
      const float* kp = kbase + (size_t)(mt + laneLo) * HD_SZ + 2 * laneHi;
#pragma unroll
      for (int ks = 0; ks < 16; ++ks) {
        v2f kf = *(const v2f*)(kp + 4 * ks);
        s = wmma4(qf[ks], kf, s);
      }
#pragma unroll
      for (int v = 0; v < 8; ++v) {
        const float pval = __expf(s[v] - rm[v]) * rinv[v];
        const int row = v + 8 * laneHi;
        const int col = (t * 16 + laneLo) * H_SZ + h;
        p_lds[row * PSTRIDE + col] = pval;
      }
    }
    __syncthreads();

    // O(16x64) += P(16xCHUNK) @ V(CHUNKx64)
#pragma unroll
    for (int ks = 0; ks < CHUNK / 4; ++ks) {
      const int k0 = 4 * ks + 2 * laneHi;
      v2f pa;
      pa.x = p_lds[laneLo * PSTRIDE + (k0    ) * H_SZ + h];
      pa.y = p_lds[laneLo * PSTRIDE + (k0 + 1) * H_SZ + h];
      const float* vp = vbase + (size_t)(m0 + k0) * HD_SZ + laneLo;
#pragma unroll
      for (int j = 0; j < 4; ++j) {
        v2f bf;
        bf.x = vp[j * 16];
        bf.y = vp[j * 16 + HD_SZ];
        oacc[j] = wmma4(pa, bf, oacc[j]);
      }
    }

    // coalesced attn write: (m,h) contiguous per query row, float4 stores
    for (int idx = threadIdx.x; idx < 16 * (CHUNK * H_SZ / 4); idx += 512) {
      const int row  = idx >> 7;      // 128 float4 per row
      const int col4 = idx & 127;
      float4 val = *(const float4*)&p_lds[row * PSTRIDE + col4 * 4];
      const size_t g = (((size_t)b * N_SZ + n0 + row) * M_SZ + m0) * H_SZ + col4 * 4;
      *(float4*)&attn[g] = val;
    }
    __syncthreads();
  }

  // write per-head output tile: (B,N,H,64)
#pragma unroll
  for (int j = 0; j < 4; ++j)
#pragma unroll
    for (int v = 0; v < 8; ++v) {
      const int row = v + 8 * laneHi;
      const int d   = j * 16 + laneLo;
      Ohead[(((size_t)b * N_SZ + (n0 + row)) * H_SZ + h) * HD_SZ + d] = oacc[j][v];
    }
}

// ---------------------------------------------------------------------------
extern "C" void kernel_launch(void* const* d_in, const int* in_sizes, int n_in,
                              void* d_out, int out_size, void* d_ws, size_t ws_size,
                              hipStream_t stream) {
  (void)in_sizes; (void)n_in; (void)out_size; (void)ws_size;
  const float* x   = (const float*)d_in[0];
  const float* y   = (const float*)d_in[1];
  const float* Wq  = (const float*)d_in[2];
  const float* bq  = (const float*)d_in[3];
  const float* Wkv = (const float*)d_in[4];
  const float* bkv = (const float*)d_in[5];
  const float* Wp  = (const float*)d_in[6];
  const float* bp  = (const float*)d_in[7];

  float* out  = (float*)d_out;
  float* attn = out + (size_t)B_SZ * N_SZ * C_SZ;

  float* q_ws  = (float*)d_ws;                          // (B,H,N,64)
  float* k_ws  = q_ws + (size_t)B_SZ * N_SZ * C_SZ;     // (B,H,M,64)
  float* v_ws  = k_ws + (size_t)B_SZ * M_SZ * C_SZ;     // (B,H,M,64)
  float* oh_ws = v_ws + (size_t)B_SZ * M_SZ * C_SZ;     // (B,N,H,64) == (B,N,C)

  { // q = x @ Wq + bq
    const int waves = ((B_SZ * N_SZ) / 32) * (C_SZ / 64);          // 2048
    gemm_wmma_kernel<<<waves / 8, 256, 0, stream>>>(x, Wq, bq, q_ws, nullptr,
                                                    C_SZ, C_SZ, N_SZ, 0);
  }
  { // kv = y @ Wkv + bkv, split into k/v
    const int waves = ((B_SZ * M_SZ) / 32) * ((2 * C_SZ) / 64);    // 4096
    gemm_wmma_kernel<<<waves / 8, 256, 0, stream>>>(y, Wkv, bkv, k_ws, v_ws,
                                                    C_SZ, 2 * C_SZ, M_SZ, 1);
  }
  attn_wmma_kernel<<<B_SZ * (N_SZ / 16), 512, 0, stream>>>(q_ws, k_ws, v_ws,
                                                           attn, oh_ws);
  { // out = ohead @ Wp + bp
    const int waves = ((B_SZ * N_SZ) / 32) * (C_SZ / 64);          // 2048
    gemm_wmma_kernel<<<waves / 8, 256, 0, stream>>>(oh_ws, Wp, bp, out, nullptr,
                                                    C_SZ, C_SZ, N_SZ, 2);
  }
}